// LennardJones_72988674228250
// MI455X (gfx1250) — compile-verified
//
#include <hip/hip_runtime.h>

// ---------------------------------------------------------------------------
// Lennard-Jones per-edge energy with scatter-add to atoms (MI455X / gfx1250).
// Memory-bound: ~0.45 GB traffic vs 0.5 GFLOP -> optimize data movement only.
// CDNA5 paths used:
//   * global_load_async_to_lds_b128 (+ s_wait_asynccnt) stages the 16-entry
//     species-pair LUT into LDS (async-tensor path, ISA ch.08)
//   * ds_load_b128 per-lane LUT lookups
//   * non-returning global_atomic_add_f32 scatter (STOREcnt, no return data)
//   * TH=NT on streamed single-use edge data so the 192MB L2 keeps the
//     reused 2MB species table + 2MB atomic accumulator resident
// ---------------------------------------------------------------------------

__global__ void zero_out(float* __restrict__ out, int n) {
    int i = blockIdx.x * blockDim.x + threadIdx.x;
    if (i < n) __builtin_nontemporal_store(0.0f, out + i);
}

// Collapse sigma/epsilon (4x4 each) into per-pair {A, B, shift, 0}:
//   A = 4*eps*s^12, B = 4*eps*s^6, shift = 4*eps*((s/rc)^12 - (s/rc)^6)
__global__ void build_lut(const float* __restrict__ sigma,
                          const float* __restrict__ epsilon,
                          float4* __restrict__ lut) {
    int t = threadIdx.x;       // 0..15 -> (si,sj) pair
    if (t < 16) {
        float s   = sigma[t];
        float e4  = 4.0f * epsilon[t];
        float s3  = s * s * s;
        float s6  = s3 * s3;
        float sc  = s * 0.2f;  // s / CUTOFF, CUTOFF = 5.0
        float sc3 = sc * sc * sc;
        float sc6 = sc3 * sc3;
        lut[t] = make_float4(e4 * s6 * s6,            // A
                             e4 * s6,                 // B
                             e4 * (sc6 * sc6 - sc6),  // shift
                             0.0f);
    }
}

__global__ void __launch_bounds__(256) lj_edges(
    const int*   __restrict__ species,
    const int*   __restrict__ first_atom,
    const int*   __restrict__ second_atom,
    const float* __restrict__ dist,     // [E,3]
    const float4* __restrict__ lut_g,   // 16 entries in d_ws
    float*       __restrict__ out,      // [N_ATOMS]
    int nedges)
{
    __shared__ float4 lut[16];

    // Stage the 256-byte LUT into LDS with the CDNA5 async-to-LDS path.
    // Lanes 0..15 of wave 0 each move 16 bytes. The flat shared address
    // carries the wave-relative LDS byte offset in its low 32 bits.
    if (threadIdx.x < 16) {
        const float4* src = lut_g + threadIdx.x;
        unsigned lds_off = (unsigned)(size_t)(&lut[threadIdx.x]);
        asm volatile("global_load_async_to_lds_b128 %0, %1, off"
                     :: "v"(lds_off), "v"(src) : "memory");
    }
    asm volatile("s_wait_asynccnt 0x0" ::: "memory");
    __syncthreads();

    int e = blockIdx.x * blockDim.x + threadIdx.x;
    if (e >= nedges) return;

    // Single-use streamed data -> non-temporal loads (don't thrash L2).
    int i  = __builtin_nontemporal_load(first_atom + e);
    int j  = __builtin_nontemporal_load(second_atom + e);
    int si = species[i];          // reused 2MB table: normal (RT) gathers, L2-resident
    int sj = species[j];

    float4 p = lut[(si << 2) | sj];   // ds_load_b128

    // Coalesced distance load (adjacent NT components, mergeable to b96)
    float dx = __builtin_nontemporal_load(dist + 3 * e + 0);
    float dy = __builtin_nontemporal_load(dist + 3 * e + 1);
    float dz = __builtin_nontemporal_load(dist + 3 * e + 2);

    float r2   = fmaf(dx, dx, fmaf(dy, dy, dz * dz));
    float r6   = r2 * r2 * r2;
    float inv6 = 1.0f / r6;                              // single FP divide
    float en   = fmaf(inv6, fmaf(p.x, inv6, -p.y), -p.z);
    float he   = 0.5f * en;

    // Non-returning hardware float atomics (STOREcnt path, no return data).
    const float* pi = out + i;
    const float* pj = out + j;
    asm volatile("global_atomic_add_f32 %0, %1, off" :: "v"(pi), "v"(he) : "memory");
    asm volatile("global_atomic_add_f32 %0, %1, off" :: "v"(pj), "v"(he) : "memory");
}

extern "C" void kernel_launch(void* const* d_in, const int* in_sizes, int n_in,
                              void* d_out, int out_size, void* d_ws, size_t ws_size,
                              hipStream_t stream) {
    const int*   species     = (const int*)  d_in[0];
    const int*   first_atom  = (const int*)  d_in[1];
    const int*   second_atom = (const int*)  d_in[2];
    const float* distances   = (const float*)d_in[3];
    const float* sigma       = (const float*)d_in[4];
    const float* epsilon     = (const float*)d_in[5];

    float*  out = (float*)d_out;
    float4* lut = (float4*)d_ws;     // 16 * 16B = 256B scratch

    int nedges = in_sizes[1];

    // 1) zero the accumulator (harness poisons d_out)
    zero_out<<<(out_size + 255) / 256, 256, 0, stream>>>(out, out_size);
    // 2) build the 16-entry species-pair LUT
    build_lut<<<1, 16, 0, stream>>>(sigma, epsilon, lut);
    // 3) main edge pass: 1 thread/edge, 8 wave32 per block
    lj_edges<<<(nedges + 255) / 256, 256, 0, stream>>>(
        species, first_atom, second_atom, distances, lut, out, nedges);
}